// CorrelationLayer_180388627250
// MI455X (gfx1250) — compile-verified
//
#include <hip/hip_runtime.h>

typedef __attribute__((ext_vector_type(2))) float v2f;
typedef __attribute__((ext_vector_type(4))) float v4f;
typedef __attribute__((ext_vector_type(8))) float v8f;

// Tensor geometry (fixed by the reference)
//   f1,f2 : (4, 64, 96, 96, 8) f32   alpha : (4, 1, 96, 96, 8) f32
//   out   : cost (4,243,96,96,8) ++ mask (4,243,96,96,8)

__launch_bounds__(128)
__global__ void corr_volume_kernel(const float* __restrict__ f1,
                                   const float* __restrict__ a1,
                                   const float* __restrict__ f2,
                                   const float* __restrict__ a2,
                                   float* __restrict__ outC,
                                   float* __restrict__ outM)
{
    // LDS staging. Layout [c>>2][d][w or col][c&3] so a K-pair (ks,ks+1) for a
    // lane is one aligned 8B ds_load; 32 lanes touch all 64 banks.
    __shared__ float lf1[16 * 8 * 16 * 4];   // 32 KB : f1 tile  [c4][d][m][ksub]
    __shared__ float lf2[16 * 8 * 24 * 4];   // 48 KB : f2 row   [c4][dp][col][ksub]
    __shared__ float lm1[16];
    __shared__ float lm2[24];

    const int bid = blockIdx.x;
    const int ii  = bid % 9;
    const int wt  = (bid / 9) % 6;
    const int h   = (bid / 54) % 96;
    const int b   = bid / (54 * 96);
    const int w0  = wt * 16;
    const int hp  = h + ii - 4;                      // shifted h row
    const bool hvalid = (hp >= 0) && (hp < 96);

    const int tid  = threadIdx.x;
    const int lane = tid & 31;
    const int wave = tid >> 5;

    const float* f1b = f1 + (size_t)b * (64 * 96 * 96 * 8);
    const float* f2b = f2 + (size_t)b * (64 * 96 * 96 * 8);

    // ---- stage f1[b, :, h, w0..w0+15, :] -> lf1[c4][d][m][ksub]
    for (int idx = tid; idx < 2048; idx += 128) {
        const int c  = idx >> 5;
        const int r  = idx & 31;
        const int w  = r >> 1;
        const int dh = (r & 1) * 4;
        const v4f v = *(const v4f*)(f1b + ((size_t)c * 96 + h) * 768 + (w0 + w) * 8 + dh);
        const int c4 = c >> 2, ksub = c & 3;
#pragma unroll
        for (int j = 0; j < 4; ++j)
            lf1[(((c4 * 8 + dh + j) * 16 + w) << 2) + ksub] = v[j];
    }

    // ---- stage f2[b, :, hp, w0-4..w0+19, :] (zero pad) -> lf2[c4][dp][col][ksub]
    for (int idx = tid; idx < 3072; idx += 128) {
        const int c   = idx / 48;
        const int r   = idx % 48;
        const int col = r >> 1;
        const int dh  = (r & 1) * 4;
        const int wp  = w0 - 4 + col;
        v4f v = {0.f, 0.f, 0.f, 0.f};
        if (hvalid && wp >= 0 && wp < 96)
            v = *(const v4f*)(f2b + ((size_t)c * 96 + hp) * 768 + wp * 8 + dh);
        const int c4 = c >> 2, ksub = c & 3;
#pragma unroll
        for (int j = 0; j < 4; ++j)
            lf2[(((c4 * 8 + dh + j) * 24 + col) << 2) + ksub] = v[j];
    }

    // ---- masks: m1 per output w, m2 per shifted column (1.0 outside image)
    if (tid < 16) {
        const float* p = a1 + ((size_t)(b * 96 + h) * 96 + (w0 + tid)) * 8;
        float s = 0.f;
#pragma unroll
        for (int j = 0; j < 8; ++j) s += p[j];
        lm1[tid] = fminf(s, 1.0f);
    } else if (tid < 40) {
        const int col = tid - 16;
        const int wp  = w0 - 4 + col;
        float val = 1.0f;                            // pad value for mask is 1.0
        if (hvalid && wp >= 0 && wp < 96) {
            const float* p = a2 + ((size_t)(b * 96 + hp) * 96 + wp) * 8;
            float s = 0.f;
#pragma unroll
            for (int j = 0; j < 8; ++j) s += p[j];
            val = fminf(s, 1.0f);
        }
        lm2[col] = val;
    }

    __syncthreads();

    // A fragment (16x4 f32): lane l<16 holds M=l, K={0,1}; l>=16 holds M=l-16, K={2,3}.
    // B fragment (4x16 f32): lane l<16 holds N=l, K={0,1}; l>=16 holds N=l-16, K={2,3}.
    const int  m  = lane & 15;
    const int  ks = (lane >> 4) << 1;                // 0 or 2
    const bool hi = lane >= 16;
    const int  n  = m;

    // 24 (kk,d) items, 6 per wave; kk/d are wave-uniform so EXEC==all-1s at WMMA.
    for (int t = 0; t < 6; ++t) {
        const int item = wave * 6 + t;
        const int kk = item >> 3;                    // 0..2  (d-shift index)
        const int d  = item & 7;
        const int dp = d + kk - 1;                   // shifted d
        const bool dvalid = (dp >= 0) && (dp < 8);

        v8f acc0 = {};                               // cols w0-4 .. w0+11
        v8f acc1 = {};                               // cols w0+4 .. w0+19
        if (dvalid) {
            for (int c4 = 0; c4 < 16; ++c4) {
                const v2f a  = *(const v2f*)(&lf1[(((c4 * 8 + d ) * 16 + m    ) << 2) + ks]);
                const v2f b0 = *(const v2f*)(&lf2[(((c4 * 8 + dp) * 24 + n    ) << 2) + ks]);
                const v2f b1 = *(const v2f*)(&lf2[(((c4 * 8 + dp) * 24 + n + 8) << 2) + ks]);
                acc0 = __builtin_amdgcn_wmma_f32_16x16x4_f32(
                           false, a, false, b0, (short)0, acc0, false, false);
                acc1 = __builtin_amdgcn_wmma_f32_16x16x4_f32(
                           false, a, false, b1, (short)0, acc1, false, false);
            }
        }

        // Band extraction: tile0 serves rows 0-7 (lanes 0-15), tile1 rows 8-15
        // (lanes 16-31). C/D layout: VGPR v -> M = v + 8*hi, N = lane%16.
        const v8f   acc     = hi ? acc1 : acc0;
        const float colMask = dvalid ? lm2[n + (hi ? 8 : 0)] : 1.0f;
        const int   Mb      = hi ? 8 : 0;
#pragma unroll
        for (int v = 0; v < 8; ++v) {
            const int jj = n - v;                    // col = w + (jj-4)
            if (jj < 0 || jj > 8) continue;
            const int   M    = Mb + v;
            const float cost = dvalid ? acc[v] * 0.015625f : 0.0f;  // mean over 64 ch
            const float cm   = fminf(lm1[M] * colMask, 1.0f);
            const int kidx = kk * 81 + ii * 9 + jj;
            const size_t off =
                ((((size_t)b * 243 + kidx) * 96 + h) * 96 + (w0 + M)) * 8 + d;
            __builtin_nontemporal_store(cost, outC + off);   // 573MB never re-read
            __builtin_nontemporal_store(cm,   outM + off);
        }
    }
}

extern "C" void kernel_launch(void* const* d_in, const int* in_sizes, int n_in,
                              void* d_out, int out_size, void* d_ws, size_t ws_size,
                              hipStream_t stream) {
    const float* f1 = (const float*)d_in[0];
    const float* a1 = (const float*)d_in[1];
    const float* f2 = (const float*)d_in[2];
    const float* a2 = (const float*)d_in[3];
    float* outC = (float*)d_out;
    float* outM = outC + (size_t)4 * 243 * 96 * 96 * 8;

    dim3 grid(4 * 96 * 6 * 9);   // (b, h, w-tile, ii)
    dim3 block(128);             // 4 waves
    hipLaunchKernelGGL(corr_volume_kernel, grid, block, 0, stream,
                       f1, a1, f2, a2, outC, outM);
}